// GAT_33938831573039
// MI455X (gfx1250) — compile-verified
//
#include <hip/hip_runtime.h>

typedef __attribute__((ext_vector_type(2))) float v2f;
typedef __attribute__((ext_vector_type(8))) float v8f;

#define NNODES 30000
#define NEDGES 480000
#define ETOT   (NEDGES + NNODES)

__device__ __forceinline__ float atomicMaxF(float* addr, float v) {
    // works for all finite/-inf values: int compare for >=0, uint-min for <0
    if (v >= 0.0f)
        return __int_as_float(atomicMax((int*)addr, __float_as_int(v)));
    else
        return __uint_as_float(atomicMin((unsigned int*)addr, __float_as_uint(v)));
}

__device__ __forceinline__ void edge_sd(const long long* __restrict__ ei, int e,
                                        int& s, int& d) {
    if (e < NEDGES) { s = (int)ei[e]; d = (int)ei[(size_t)NEDGES + e]; }
    else            { s = e - NEDGES; d = s; }   // self-loops appended
}

// ---------------------------------------------------------------------------
// C[M,Nc] = A[M,K] * B[K,Nc], fp32. One wave computes a 16x32 output strip
// (two 16x16 WMMA tiles sharing the A fragment) via V_WMMA_F32_16X16X4_F32.
// A layout: lane(0..31): M = lane&15, k = k0 + 2*(lane>>4) + j  (j = vec elem)
// B layout: lane: N = lane&15, k = k0 + 2*(lane>>4) + j
// C layout: vgpr r, lane: M = r + 8*(lane>>4), N = lane&15
// K tail (K=129) handled branch-free: clamped address + mask multiply.
// ---------------------------------------------------------------------------
__global__ void gemm_f32_wmma(const float* __restrict__ A,
                              const float* __restrict__ B,
                              float* __restrict__ C,
                              int K, int Nc) {
    const int lane = threadIdx.x & 31;
    const int half = lane >> 4;
    const int nl   = lane & 15;
    const int m0   = blockIdx.x * 16;
    const int n0   = blockIdx.y * 32;
    const int n1   = n0 + 16;
    const bool has2 = (n1 < Nc);
    const int n1c  = has2 ? n1 : 0;          // clamp for safe (discarded) loads

    v8f acc0 = {}, acc1 = {};
    const float* Arow = A + (size_t)(m0 + nl) * K;
    const int K4 = K & ~3;

#pragma unroll 2
    for (int k0 = 0; k0 < K4; k0 += 4) {
        const int ka = k0 + 2 * half;
        v2f a, b0, b1;
        a[0] = Arow[ka];
        a[1] = Arow[ka + 1];
        const float* Brow0 = B + (size_t)(ka + 0) * Nc;
        const float* Brow1 = B + (size_t)(ka + 1) * Nc;
        b0[0] = Brow0[n0 + nl];  b0[1] = Brow1[n0 + nl];
        b1[0] = Brow0[n1c + nl]; b1[1] = Brow1[n1c + nl];
        acc0 = __builtin_amdgcn_wmma_f32_16x16x4_f32(false, a, false, b0,
                                                     (short)0, acc0, false, false);
        acc1 = __builtin_amdgcn_wmma_f32_16x16x4_f32(false, a, false, b1,
                                                     (short)0, acc1, false, false);
    }

    if (K4 < K) {                             // only layer 1 (K=129) takes this
        const int   ka  = K4 + 2 * half;
        const float w0  = (ka + 0 < K) ? 1.0f : 0.0f;
        const float w1  = (ka + 1 < K) ? 1.0f : 0.0f;
        const int   ka0 = (ka + 0 < K) ? ka + 0 : (K - 1);
        const int   ka1 = (ka + 1 < K) ? ka + 1 : (K - 1);
        v2f a, b0, b1;
        a[0] = Arow[ka0] * w0;
        a[1] = Arow[ka1] * w1;
        const float* Brow0 = B + (size_t)ka0 * Nc;
        const float* Brow1 = B + (size_t)ka1 * Nc;
        b0[0] = Brow0[n0 + nl] * w0;  b0[1] = Brow1[n0 + nl] * w1;
        b1[0] = Brow0[n1c + nl] * w0; b1[1] = Brow1[n1c + nl] * w1;
        acc0 = __builtin_amdgcn_wmma_f32_16x16x4_f32(false, a, false, b0,
                                                     (short)0, acc0, false, false);
        acc1 = __builtin_amdgcn_wmma_f32_16x16x4_f32(false, a, false, b1,
                                                     (short)0, acc1, false, false);
    }

#pragma unroll
    for (int r = 0; r < 8; ++r)
        C[(size_t)(m0 + r + 8 * half) * Nc + n0 + nl] = acc0[r];
    if (has2) {                               // wave-uniform branch, EXEC intact
#pragma unroll
        for (int r = 0; r < 8; ++r)
            C[(size_t)(m0 + r + 8 * half) * Nc + n1 + nl] = acc1[r];
    }
}

// alpha_s[n,h] = sum_c h[n,h*C+c]*a_s[h,c] ; likewise alpha_d. One wave per (n,h).
__global__ void alpha_kernel(const float* __restrict__ h,
                             const float* __restrict__ a_s,
                             const float* __restrict__ a_d,
                             float* __restrict__ out_s,
                             float* __restrict__ out_d,
                             int H, int C) {
    const int gw   = (blockIdx.x * blockDim.x + threadIdx.x) >> 5;
    const int lane = threadIdx.x & 31;
    if (gw >= NNODES * H) return;
    const int n  = gw / H;
    const int hh = gw - n * H;
    const int HC = H * C;
    float s = 0.0f, d = 0.0f;
    for (int c = lane; c < C; c += 32) {
        const float v = h[(size_t)n * HC + hh * C + c];
        s += v * a_s[hh * C + c];
        d += v * a_d[hh * C + c];
    }
#pragma unroll
    for (int off = 16; off > 0; off >>= 1) {
        s += __shfl_down(s, off, 32);
        d += __shfl_down(d, off, 32);
    }
    if (lane == 0) { out_s[gw] = s; out_d[gw] = d; }
}

// init m=-inf, denom=0, out = broadcast(bias)
__global__ void init_nodes(float* __restrict__ m, float* __restrict__ den,
                           float* __restrict__ out, const float* __restrict__ bias,
                           int H, int HC) {
    const long long i = (long long)blockIdx.x * blockDim.x + threadIdx.x;
    if (i < (long long)NNODES * H) { m[i] = -__builtin_inff(); den[i] = 0.0f; }
    if (i < (long long)NNODES * HC) out[i] = bias[(int)(i % HC)];
}

__global__ void edge_max_kernel(const long long* __restrict__ ei,
                                const float* __restrict__ as,
                                const float* __restrict__ ad,
                                float* __restrict__ m, int H) {
    const int e = blockIdx.x * blockDim.x + threadIdx.x;
    if (e >= ETOT) return;
    int s, d; edge_sd(ei, e, s, d);
    for (int h = 0; h < H; ++h) {
        float v = as[s * H + h] + ad[d * H + h];
        v = v > 0.0f ? v : 0.2f * v;                 // leaky_relu(0.2)
        atomicMaxF(&m[d * H + h], v);
    }
}

__global__ void edge_expsum_kernel(const long long* __restrict__ ei,
                                   const float* __restrict__ as,
                                   const float* __restrict__ ad,
                                   const float* __restrict__ m,
                                   float* __restrict__ den,
                                   float* __restrict__ ex, int H) {
    const int e = blockIdx.x * blockDim.x + threadIdx.x;
    if (e >= ETOT) return;
    int s, d; edge_sd(ei, e, s, d);
    for (int h = 0; h < H; ++h) {
        float v = as[s * H + h] + ad[d * H + h];
        v = v > 0.0f ? v : 0.2f * v;
        const float x = __expf(v - m[d * H + h]);
        ex[(size_t)e * H + h] = x;
        atomicAdd(&den[d * H + h], x);
    }
}

// one wave per edge: out[dst] += h[src] * (ex/denom[dst])  (L2-resident)
__global__ void edge_agg_kernel(const long long* __restrict__ ei,
                                const float* __restrict__ h,
                                const float* __restrict__ ex,
                                const float* __restrict__ den,
                                float* __restrict__ out, int H, int C) {
    const int we   = (blockIdx.x * blockDim.x + threadIdx.x) >> 5;
    const int lane = threadIdx.x & 31;
    if (we >= ETOT) return;
    int s, d; edge_sd(ei, we, s, d);
    const int HC = H * C;
    for (int hh = 0; hh < H; ++hh) {
        const float w = ex[(size_t)we * H + hh] / (den[d * H + hh] + 1e-16f);
        const float* hs = h   + (size_t)s * HC + hh * C;
        float*       od = out + (size_t)d * HC + hh * C;
        for (int c = lane; c < C; c += 32)
            atomicAdd(&od[c], hs[c] * w);
    }
}

__global__ void relu_kernel(float* __restrict__ x, long long n) {
    const long long i = (long long)blockIdx.x * blockDim.x + threadIdx.x;
    if (i < n) x[i] = fmaxf(x[i], 0.0f);
}

extern "C" void kernel_launch(void* const* d_in, const int* in_sizes, int n_in,
                              void* d_out, int out_size, void* d_ws, size_t ws_size,
                              hipStream_t stream) {
    const float*     x   = (const float*)d_in[0];
    const long long* ei  = (const long long*)d_in[1];
    const float* W1  = (const float*)d_in[2];
    const float* a1s = (const float*)d_in[3];
    const float* a1d = (const float*)d_in[4];
    const float* b1  = (const float*)d_in[5];
    const float* W2  = (const float*)d_in[6];
    const float* a2s = (const float*)d_in[7];
    const float* a2d = (const float*)d_in[8];
    const float* b2  = (const float*)d_in[9];
    const float* W3  = (const float*)d_in[10];
    const float* a3s = (const float*)d_in[11];
    const float* a3d = (const float*)d_in[12];
    const float* b3  = (const float*)d_in[13];
    float* out = (float*)d_out;

    // workspace layout (floats)
    float* hbuf = (float*)d_ws;                         // [30000, 448] pre-agg h
    float* obuf = hbuf + (size_t)NNODES * 448;          // [30000, 448] post-agg
    float* asb  = obuf + (size_t)NNODES * 448;          // [30000, 7]
    float* adb  = asb  + (size_t)NNODES * 7;
    float* mb   = adb  + (size_t)NNODES * 7;
    float* db   = mb   + (size_t)NNODES * 7;
    float* exb  = db   + (size_t)NNODES * 7;            // [510000, 7]

    auto layer = [&](const float* in, const float* W, const float* avs,
                     const float* avd, const float* bias, float* hout,
                     float* oout, int K, int H, int C, bool do_relu) {
        const int HC = H * C;
        dim3 g1(NNODES / 16, (HC / 16 + 1) / 2);        // 16x32 strip per wave
        gemm_f32_wmma<<<g1, 32, 0, stream>>>(in, W, hout, K, HC);

        {
            long long thr = (long long)NNODES * H * 32;
            alpha_kernel<<<(unsigned)((thr + 255) / 256), 256, 0, stream>>>(
                hout, avs, avd, asb, adb, H, C);
        }
        {
            long long tot = (long long)NNODES * HC;
            init_nodes<<<(unsigned)((tot + 255) / 256), 256, 0, stream>>>(
                mb, db, oout, bias, H, HC);
        }
        edge_max_kernel<<<(ETOT + 255) / 256, 256, 0, stream>>>(ei, asb, adb, mb, H);
        edge_expsum_kernel<<<(ETOT + 255) / 256, 256, 0, stream>>>(ei, asb, adb, mb,
                                                                   db, exb, H);
        {
            long long thr = (long long)ETOT * 32;
            edge_agg_kernel<<<(unsigned)((thr + 255) / 256), 256, 0, stream>>>(
                ei, hout, exb, db, oout, H, C);
        }
        if (do_relu) {
            long long tot = (long long)NNODES * HC;
            relu_kernel<<<(unsigned)((tot + 255) / 256), 256, 0, stream>>>(oout, tot);
        }
    };

    // layer 1: in=129, H=7, C=64  (K tail handled branch-free)
    layer(x,    W1, a1s, a1d, b1, hbuf, obuf, 129, 7, 64, true);
    // layer 2: in=448, H=6, C=64   (obuf consumed by GEMM before init overwrites it)
    layer(obuf, W2, a2s, a2d, b2, hbuf, obuf, 448, 6, 64, true);
    // layer 3: in=384, H=6, C=40 -> d_out
    layer(obuf, W3, a3s, a3d, b3, hbuf, out, 384, 6, 40, false);
}